// NonMaxSuppression_11261404250300
// MI455X (gfx1250) — compile-verified
//
#include <hip/hip_runtime.h>
#include <hip/hip_bf16.h>
#include <stdint.h>

// NMS over [16,1,1024,1024] f32, then order-preserving compaction of (y,x)
// coords of maxima >= 0.7 into int32 out[2][K], zero-padded.
//
// Bandwidth-bound (~100 MB total traffic -> ~5-10us at 23.3 TB/s). Input rows
// are staged global->LDS with gfx1250 async copies (ASYNCcnt path) so the 3x3
// window is computed out of LDS with no VGPR staging.

#define REP_THR  0.7f
#define IMG_W    1024
#define IMG_H    1024
#define RSTRIDE  1028                 // floats; 1028*4 bytes, keeps 16B align
#define NEGINF   (-__builtin_inff())

typedef __attribute__((ext_vector_type(4))) float f4;
typedef int v4i __attribute__((vector_size(16)));
typedef __attribute__((address_space(1))) v4i* g_v4i_ptr;   // global int4*
typedef __attribute__((address_space(3))) v4i* l_v4i_ptr;   // LDS int4*

#if defined(__has_builtin)
#if __has_builtin(__builtin_amdgcn_global_load_async_to_lds_b128)
#define USE_ASYNC_COPY 1
#endif
#endif

__device__ __forceinline__ void g2l_b128(float* lds_dst, const float* g_src) {
#if defined(USE_ASYNC_COPY)
  // global_load_async_to_lds_b128: per-lane 16B global -> LDS, ASYNCcnt-tracked
  __builtin_amdgcn_global_load_async_to_lds_b128(
      (g_v4i_ptr)g_src, (l_v4i_ptr)lds_dst, /*imm offset*/0, /*cpol*/0);
#else
  *(f4*)lds_dst = *(const f4*)g_src;
#endif
}

__device__ __forceinline__ void async_wait_all() {
#if defined(USE_ASYNC_COPY)
#if __has_builtin(__builtin_amdgcn_s_wait_asynccnt)
  __builtin_amdgcn_s_wait_asynccnt(0);
#else
  asm volatile("s_wait_asynccnt 0" ::: "memory");
#endif
#endif
}

__device__ __forceinline__ unsigned ballot32(bool p) {
#if defined(__has_builtin) && __has_builtin(__builtin_amdgcn_ballot_w32)
  return __builtin_amdgcn_ballot_w32(p);
#else
  return (unsigned)__ballot(p);
#endif
}

// Stage rows y-1,y,y+1 of image b into LDS (out-of-range rows filled -inf).
__device__ __forceinline__ void load_rows(float* smem, const float* img,
                                          int b, int y, int tid) {
  const int p = tid * 4;  // this thread's 4 columns
#pragma unroll
  for (int k = 0; k < 3; ++k) {
    const int ys = y - 1 + k;
    float* dst = &smem[k * RSTRIDE + p];
    if (ys >= 0 && ys < IMG_H) {
      const float* src = img + ((size_t)b * IMG_H + ys) * IMG_W + p;
      g2l_b128(dst, src);
    } else {
      f4 v = {NEGINF, NEGINF, NEGINF, NEGINF};
      *(f4*)dst = v;
    }
  }
  async_wait_all();
  __syncthreads();
}

// 4-bit mask: bit j set if pixel (y, tid*4+j) is a 3x3 local max >= REP_THR.
__device__ __forceinline__ unsigned nms_mask_row(const float* smem, int tid) {
  const int p = tid * 4;
  const f4 a0 = *(const f4*)&smem[0 * RSTRIDE + p];
  const f4 a1 = *(const f4*)&smem[1 * RSTRIDE + p];
  const f4 a2 = *(const f4*)&smem[2 * RSTRIDE + p];
  // x-halo: -inf outside the image (p==0 only for tid 0, p+4==W only for last)
  const float L0 = (p == 0) ? NEGINF : smem[0 * RSTRIDE + p - 1];
  const float L1 = (p == 0) ? NEGINF : smem[1 * RSTRIDE + p - 1];
  const float L2 = (p == 0) ? NEGINF : smem[2 * RSTRIDE + p - 1];
  const float R0 = (p + 4 == IMG_W) ? NEGINF : smem[0 * RSTRIDE + p + 4];
  const float R1 = (p + 4 == IMG_W) ? NEGINF : smem[1 * RSTRIDE + p + 4];
  const float R2 = (p + 4 == IMG_W) ? NEGINF : smem[2 * RSTRIDE + p + 4];

  const float vL = fmaxf(fmaxf(L0, L1), L2);           // vertical maxima
  const float v0 = fmaxf(fmaxf(a0.x, a1.x), a2.x);
  const float v1 = fmaxf(fmaxf(a0.y, a1.y), a2.y);
  const float v2 = fmaxf(fmaxf(a0.z, a1.z), a2.z);
  const float v3 = fmaxf(fmaxf(a0.w, a1.w), a2.w);
  const float vR = fmaxf(fmaxf(R0, R1), R2);

  unsigned m = 0;
  const float p0 = fmaxf(vL, fmaxf(v0, v1));
  const float p1 = fmaxf(v0, fmaxf(v1, v2));
  const float p2 = fmaxf(v1, fmaxf(v2, v3));
  const float p3 = fmaxf(v2, fmaxf(v3, vR));
  if (a1.x >= REP_THR && a1.x == p0) m |= 1u;
  if (a1.y >= REP_THR && a1.y == p1) m |= 2u;
  if (a1.z >= REP_THR && a1.z == p2) m |= 4u;
  if (a1.w >= REP_THR && a1.w == p3) m |= 8u;
  return m;
}

// ---------------- Pass 0: zero-fill output (padding fill_value = 0) --------
// b128 stores: full-cacheline zeroing at HBM rate; scalar tail for n % 4.
__global__ void zero_kernel(int* __restrict__ out, int n) {
  const int i = blockIdx.x * blockDim.x + threadIdx.x;
  const int n4 = n >> 2;
  if (i < n4) {
    v4i z = {0, 0, 0, 0};
    *(v4i*)(out + i * 4) = z;
  }
  const int tail = n - n4 * 4;               // 0..3 leftover ints
  if (i < tail) out[n4 * 4 + i] = 0;
}

// ---------------- Pass 1: per-row maxima counts ----------------------------
__global__ __launch_bounds__(256) void nms_count_kernel(
    const float* __restrict__ in, int* __restrict__ counts) {
  __shared__ __align__(16) float smem[3 * RSTRIDE];
  __shared__ int wsum[8];
  const int tid = threadIdx.x;
  const int r = blockIdx.x;          // flat row index: r = b*H + y
  const int b = r >> 10;
  const int y = r & (IMG_H - 1);

  load_rows(smem, in, b, y, tid);
  const unsigned m = nms_mask_row(smem, tid);

  const unsigned b0 = ballot32((m & 1u) != 0);
  const unsigned b1 = ballot32((m & 2u) != 0);
  const unsigned b2 = ballot32((m & 4u) != 0);
  const unsigned b3 = ballot32((m & 8u) != 0);
  const int wave = tid >> 5, lane = tid & 31;
  const int wc = __builtin_popcount(b0) + __builtin_popcount(b1) +
                 __builtin_popcount(b2) + __builtin_popcount(b3);
  if (lane == 0) wsum[wave] = wc;
  __syncthreads();
  if (tid == 0) {
    int s = 0;
#pragma unroll
    for (int w = 0; w < 8; ++w) s += wsum[w];
    counts[r] = s;
  }
}

// ---------------- Pass 2: exclusive scan of 16384 row counts ---------------
__global__ __launch_bounds__(1024) void scan_kernel(
    const int* __restrict__ counts, int* __restrict__ offsets, int n) {
  __shared__ int sdata[1024];
  const int tid = threadIdx.x;
  const int chunk = (n + 1023) >> 10;
  const int beg = tid * chunk;
  const int end = (beg + chunk < n) ? (beg + chunk) : n;

  int local = 0;
  for (int i = beg; i < end; ++i) local += counts[i];
  sdata[tid] = local;
  __syncthreads();
  for (int off = 1; off < 1024; off <<= 1) {   // Hillis-Steele inclusive scan
    int v = 0;
    if (tid >= off) v = sdata[tid - off];
    __syncthreads();
    sdata[tid] += v;
    __syncthreads();
  }
  int run = sdata[tid] - local;                // exclusive prefix of my chunk
  for (int i = beg; i < end; ++i) { offsets[i] = run; run += counts[i]; }
}

// ---------------- Pass 3: ordered scatter of (y, x) ------------------------
__global__ __launch_bounds__(256) void nms_scatter_kernel(
    const float* __restrict__ in, const int* __restrict__ offsets,
    int* __restrict__ out, int K) {
  __shared__ __align__(16) float smem[3 * RSTRIDE];
  __shared__ int wsum[8];
  const int tid = threadIdx.x;
  const int r = blockIdx.x;
  const int b = r >> 10;
  const int y = r & (IMG_H - 1);

  load_rows(smem, in, b, y, tid);
  const unsigned m = nms_mask_row(smem, tid);

  const unsigned b0 = ballot32((m & 1u) != 0);
  const unsigned b1 = ballot32((m & 2u) != 0);
  const unsigned b2 = ballot32((m & 4u) != 0);
  const unsigned b3 = ballot32((m & 8u) != 0);
  const int wave = tid >> 5, lane = tid & 31;
  const int wc = __builtin_popcount(b0) + __builtin_popcount(b1) +
                 __builtin_popcount(b2) + __builtin_popcount(b3);
  if (lane == 0) wsum[wave] = wc;
  __syncthreads();

  int base = offsets[r];
  for (int w = 0; w < wave; ++w) base += wsum[w];      // cross-wave rank
  const unsigned below = (1u << lane) - 1u;            // lanes before me
  const int pre = __builtin_popcount(b0 & below) + __builtin_popcount(b1 & below) +
                  __builtin_popcount(b2 & below) + __builtin_popcount(b3 & below);
  int idx = base + pre;
  const int p = tid * 4;
#pragma unroll
  for (int j = 0; j < 4; ++j) {                        // intra-lane, j order
    if (m & (1u << j)) {
      if (idx < K) { out[idx] = y; out[K + idx] = p + j; }
      ++idx;
    }
  }
}

extern "C" void kernel_launch(void* const* d_in, const int* in_sizes, int n_in,
                              void* d_out, int out_size, void* d_ws, size_t ws_size,
                              hipStream_t stream) {
  const float* in = (const float*)d_in[0];     // [B,1,H,W] f32
  int* out = (int*)d_out;                      // [2, K] int32
  const int total = in_sizes[0];               // B*H*W
  const int nrows = total / IMG_W;             // B*H = 16384
  const int K = out_size / 2;

  int* counts  = (int*)d_ws;                   // nrows ints
  int* offsets = counts + nrows;               // nrows ints (128 KB total)

  const int zthreads = out_size / 4 + 4;       // covers vec body + tail
  zero_kernel<<<(zthreads + 255) / 256, 256, 0, stream>>>(out, out_size);
  nms_count_kernel<<<nrows, 256, 0, stream>>>(in, counts);
  scan_kernel<<<1, 1024, 0, stream>>>(counts, offsets, nrows);
  nms_scatter_kernel<<<nrows, 256, 0, stream>>>(in, offsets, out, K);
}